// GCN_71536975282280
// MI455X (gfx1250) — compile-verified
//
#include <hip/hip_runtime.h>
#include <hip/hip_bf16.h>

typedef __attribute__((ext_vector_type(2))) float v2f;
typedef __attribute__((ext_vector_type(8))) float v8f;

// ---------------------------------------------------------------------------
// Degree / norm precompute
// ---------------------------------------------------------------------------
__global__ void k_init_deg(float* deg, int N) {
    int i = blockIdx.x * blockDim.x + threadIdx.x;
    if (i < N) deg[i] = 1.0f;  // self-loop weight
}

__global__ void k_deg_accum(const int* __restrict__ col, const float* __restrict__ ew,
                            float* deg, int E) {
    int e = blockIdx.x * blockDim.x + threadIdx.x;
    if (e < E) atomicAdd(&deg[col[e]], ew[e]);
}

__global__ void k_finalize_deg(float* dis, float* self_norm, int N) {
    int i = blockIdx.x * blockDim.x + threadIdx.x;
    if (i >= N) return;
    float d = dis[i];
    float r = (d > 0.0f) ? rsqrtf(d) : 0.0f;
    dis[i] = r;
    self_norm[i] = r * r;
}

__global__ void k_edge_norm(const int* __restrict__ row, const int* __restrict__ col,
                            const float* __restrict__ ew, const float* __restrict__ dis,
                            float* __restrict__ normv, int E) {
    int e = blockIdx.x * blockDim.x + threadIdx.x;
    if (e >= E) return;
    normv[e] = dis[row[e]] * ew[e] * dis[col[e]];
}

// ---------------------------------------------------------------------------
// fp32 WMMA GEMM:  out[Nrows x Fout] = act(X[Nrows x 128]) @ W[128 x Fout]
// One wave -> one 16x16 tile.  B fragments (128x16 W slab) preloaded into
// registers once per block; A staged in LDS (padded, conflict-free).
// ---------------------------------------------------------------------------
__global__ void k_gemm_wmma_f32(const float* __restrict__ X, const float* __restrict__ W,
                                float* __restrict__ out, int Nrows, int Fout, int reluIn) {
    constexpr int K   = 128;
    constexpr int LDA = K + 4;               // 132 floats -> bank-conflict-free
    __shared__ float ldsA[16 * LDA];

    const int lane  = threadIdx.x & 31;
    const int wave  = threadIdx.x >> 5;      // 0 .. Fout/16 - 1
    const int n0    = wave * 16;
    const int khalf = lane >> 4;             // 0 for lanes 0-15, 1 for 16-31
    const int l15   = lane & 15;

    // Preload B fragments: V_WMMA_F32_16X16X4_F32 B layout (4x16, 2 VGPRs):
    //   VGPR v: lanes 0-15 -> K=k0+v, N=n0+lane ; lanes 16-31 -> K=k0+v+2
    v2f bfrag[32];
#pragma unroll
    for (int s = 0; s < 32; ++s) {
        const int k0 = s * 4;
        v2f b;
        b.x = W[(size_t)(k0 + 0 + 2 * khalf) * Fout + n0 + l15];
        b.y = W[(size_t)(k0 + 1 + 2 * khalf) * Fout + n0 + l15];
        bfrag[s] = b;
    }

    const int numTiles = (Nrows + 15) >> 4;
    for (int t = blockIdx.x; t < numTiles; t += gridDim.x) {
        const int row0 = t << 4;
        __syncthreads();
        // Stage A tile [16 x 128] into LDS (fused ReLU of previous layer)
        for (int f4 = threadIdx.x; f4 < 512; f4 += blockDim.x) {
            const int r  = f4 >> 5;
            const int c4 = (f4 & 31) << 2;
            const int gr = row0 + r;
            float4 v = make_float4(0.f, 0.f, 0.f, 0.f);
            if (gr < Nrows) v = *(const float4*)(X + (size_t)gr * K + c4);
            if (reluIn) {
                v.x = fmaxf(v.x, 0.f); v.y = fmaxf(v.y, 0.f);
                v.z = fmaxf(v.z, 0.f); v.w = fmaxf(v.w, 0.f);
            }
            *(float4*)&ldsA[r * LDA + c4] = v;
        }
        __syncthreads();

        v8f c = {0.f, 0.f, 0.f, 0.f, 0.f, 0.f, 0.f, 0.f};
#pragma unroll
        for (int s = 0; s < 32; ++s) {
            const int k0 = s * 4;
            // A layout (16x4, 2 VGPRs): lanes 0-15 hold K={k0,k0+1},
            // lanes 16-31 hold K={k0+2,k0+3}, M = lane&15
            const float* ap = &ldsA[l15 * LDA + k0 + 2 * khalf];
            v2f a;
            a.x = ap[0];
            a.y = ap[1];
            c = __builtin_amdgcn_wmma_f32_16x16x4_f32(
                    false, a, false, bfrag[s], (short)0, c, false, false);
        }

        // D layout: VGPR r -> M=r (lanes 0-15), M=r+8 (lanes 16-31); N=lane&15
        const int n     = n0 + l15;
        const int mbase = row0 + (khalf ? 8 : 0);
#pragma unroll
        for (int r = 0; r < 8; ++r) {
            const int m = mbase + r;
            if (m < Nrows) out[(size_t)m * Fout + n] = c[r];
        }
    }
}

// ---------------------------------------------------------------------------
// out[n][f] = bias[f] + self_norm[n] * tmp[n][f]    (full overwrite, no memset)
// ---------------------------------------------------------------------------
__global__ void k_init_out(const float* __restrict__ tmp, const float* __restrict__ bias,
                           const float* __restrict__ self_norm, float* __restrict__ out,
                           int Nrows, int f4shift) {
    const size_t i = (size_t)blockIdx.x * blockDim.x + threadIdx.x;
    const size_t total = (size_t)Nrows << f4shift;
    if (i >= total) return;
    const size_t n  = i >> f4shift;
    const int    f4 = (int)(i & ((1u << f4shift) - 1u));
    const float4 t = ((const float4*)tmp)[i];
    const float4 b = ((const float4*)bias)[f4];
    const float  s = self_norm[n];
    float4 o;
    o.x = fmaf(s, t.x, b.x); o.y = fmaf(s, t.y, b.y);
    o.z = fmaf(s, t.z, b.z); o.w = fmaf(s, t.w, b.w);
    ((float4*)out)[i] = o;
}

// ---------------------------------------------------------------------------
// Edge scatter: one wave32 per edge; coalesced gather of h[row], f32 atomic
// scatter-add into out[col].  VEC = F/32 (4 for F=128, 2 for F=64).
// ---------------------------------------------------------------------------
template <int VEC>
__global__ void k_edge_scatter(const int* __restrict__ row, const int* __restrict__ col,
                               const float* __restrict__ normv, const float* __restrict__ tmp,
                               float* out, int E) {
    const int e    = (int)(((size_t)blockIdx.x * blockDim.x + threadIdx.x) >> 5);
    const int lane = threadIdx.x & 31;
    if (e >= E) return;
    const int   r = row[e];
    const int   c = col[e];
    const float w = normv[e];
    constexpr int F = VEC * 32;
    const float* src = tmp + (size_t)r * F + lane * VEC;
    float*       dst = out + (size_t)c * F + lane * VEC;
    if constexpr (VEC == 4) {
        const float4 v = *(const float4*)src;
        atomicAdd(dst + 0, w * v.x);
        atomicAdd(dst + 1, w * v.y);
        atomicAdd(dst + 2, w * v.z);
        atomicAdd(dst + 3, w * v.w);
    } else {
        const float2 v = *(const float2*)src;
        atomicAdd(dst + 0, w * v.x);
        atomicAdd(dst + 1, w * v.y);
    }
}

// ---------------------------------------------------------------------------
// In-place log_softmax over 64 classes; one wave32 per row, shfl reductions.
// ---------------------------------------------------------------------------
__global__ void k_logsoftmax64(float* out, int Nrows) {
    const int n    = (int)(((size_t)blockIdx.x * blockDim.x + threadIdx.x) >> 5);
    const int lane = threadIdx.x & 31;
    if (n >= Nrows) return;
    float* p = out + (size_t)n * 64;
    const float v0 = p[lane];
    const float v1 = p[lane + 32];
    float m = fmaxf(v0, v1);
#pragma unroll
    for (int off = 16; off > 0; off >>= 1) m = fmaxf(m, __shfl_xor(m, off, 32));
    float s = expf(v0 - m) + expf(v1 - m);
#pragma unroll
    for (int off = 16; off > 0; off >>= 1) s += __shfl_xor(s, off, 32);
    const float lg = m + logf(s);
    p[lane]      = v0 - lg;
    p[lane + 32] = v1 - lg;
}

// ---------------------------------------------------------------------------
// Launch
// ---------------------------------------------------------------------------
extern "C" void kernel_launch(void* const* d_in, const int* in_sizes, int n_in,
                              void* d_out, int out_size, void* d_ws, size_t ws_size,
                              hipStream_t stream) {
    const float* x  = (const float*)d_in[0];
    const int*   ei = (const int*)d_in[1];
    const float* ew = (const float*)d_in[2];
    const float* W1 = (const float*)d_in[3];
    const float* b1 = (const float*)d_in[4];
    const float* W2 = (const float*)d_in[5];
    const float* b2 = (const float*)d_in[6];
    const float* W3 = (const float*)d_in[7];
    const float* b3 = (const float*)d_in[8];

    const int H   = in_sizes[4];            // 128
    const int C   = in_sizes[8];            // 64
    const int Fin = in_sizes[3] / H;        // 128
    const int N   = in_sizes[0] / Fin;      // 100000
    const int E   = in_sizes[2];            // 1600000
    const int* row = ei;
    const int* col = ei + E;

    float* ws        = (float*)d_ws;
    float* dis       = ws;                          // N   (deg -> dis in place)
    float* self_norm = dis + N;                     // N
    float* normv     = self_norm + N;               // E
    float* A         = normv + E;                   // N*128 (gemm results)
    float* B         = A + (size_t)N * 128;         // N*128 (aggregated acts)
    float* outF      = (float*)d_out;               // N*C

    const int TB = 256;
    const int gN = (N + TB - 1) / TB;
    const int gE = (E + TB - 1) / TB;

    // --- normalization coefficients (shared by all layers) ---
    k_init_deg<<<gN, TB, 0, stream>>>(dis, N);
    k_deg_accum<<<gE, TB, 0, stream>>>(col, ew, dis, E);
    k_finalize_deg<<<gN, TB, 0, stream>>>(dis, self_norm, N);
    k_edge_norm<<<gE, TB, 0, stream>>>(row, col, ew, dis, normv, E);

    const int gemmGrid = 1024;
    const int gEdgeW   = (int)(((size_t)E * 32 + TB - 1) / TB);

    // --- layer 1: h1 = relu(agg(x @ W1)) ---  (ReLU deferred into next GEMM)
    k_gemm_wmma_f32<<<gemmGrid, dim3((H / 16) * 32), 0, stream>>>(x, W1, A, N, H, 0);
    k_init_out<<<(int)(((size_t)N * (H / 4) + TB - 1) / TB), TB, 0, stream>>>(
        A, b1, self_norm, B, N, 5);
    k_edge_scatter<4><<<gEdgeW, TB, 0, stream>>>(row, col, normv, A, B, E);

    // --- layer 2: h2 = relu(agg(relu(h1) @ W2)) ---
    k_gemm_wmma_f32<<<gemmGrid, dim3((H / 16) * 32), 0, stream>>>(B, W2, A, N, H, 1);
    k_init_out<<<(int)(((size_t)N * (H / 4) + TB - 1) / TB), TB, 0, stream>>>(
        A, b2, self_norm, B, N, 5);
    k_edge_scatter<4><<<gEdgeW, TB, 0, stream>>>(row, col, normv, A, B, E);

    // --- layer 3: out = log_softmax(agg(relu(h2) @ W3)) ---
    k_gemm_wmma_f32<<<gemmGrid, dim3((C / 16) * 32), 0, stream>>>(B, W3, A, N, C, 1);
    k_init_out<<<(int)(((size_t)N * (C / 4) + TB - 1) / TB), TB, 0, stream>>>(
        A, b3, self_norm, outF, N, 4);
    k_edge_scatter<2><<<gEdgeW, TB, 0, stream>>>(row, col, normv, A, outF, E);
    k_logsoftmax64<<<(int)(((size_t)N * 32 + TB - 1) / TB), TB, 0, stream>>>(outF, N);
}